// H2GCN_26474178413288
// MI455X (gfx1250) — compile-verified
//
#include <hip/hip_runtime.h>
#include <cstdint>

typedef __attribute__((ext_vector_type(2))) float v2f;
typedef __attribute__((ext_vector_type(8))) float v8f;
typedef int v4i __attribute__((vector_size(16)));   // matches builtin param type

// ------------------------------------------------------------------
// Elementwise / graph-normalization kernels
// ------------------------------------------------------------------
__global__ void k_fill_f32(float* __restrict__ p, float v, long n) {
  long t = (long)blockIdx.x * blockDim.x + threadIdx.x;
  if (t < n) p[t] = v;
}

__global__ void k_count_deg(const int64_t* __restrict__ col,
                            float* __restrict__ deg, long E) {
  long t = (long)blockIdx.x * blockDim.x + threadIdx.x;
  if (t < E) atomicAdd(&deg[col[t]], 1.0f);
}

__global__ void k_rsqrt_inplace(float* __restrict__ p, long n) {
  long t = (long)blockIdx.x * blockDim.x + threadIdx.x;
  if (t < n) p[t] = rsqrtf(p[t]);   // deg >= 1 always (self loops)
}

__global__ void k_edge_norm(const int64_t* __restrict__ row,
                            const int64_t* __restrict__ col,
                            const float* __restrict__ dis,
                            float* __restrict__ norm, long E) {
  long t = (long)blockIdx.x * blockDim.x + threadIdx.x;
  if (t < E) norm[t] = dis[row[t]] * dis[col[t]];
}

// zero one 64-wide column block of feat (row stride ld)
__global__ void k_zero_block(float* __restrict__ blk, int ld, long N) {
  long t = (long)blockIdx.x * blockDim.x + threadIdx.x;
  if (t >= N * 64) return;
  long n = t >> 6; int h = (int)(t & 63);
  blk[n * (long)ld + h] = 0.0f;
}

// scatter: blk[col[e]*ld + h] += norm[e] * hw[row[e]*64 + h]
// 16 lanes per edge, 4 features (float4) per lane
__global__ void k_scatter_edges(const int64_t* __restrict__ row,
                                const int64_t* __restrict__ col,
                                const float* __restrict__ norm,
                                const float* __restrict__ hw,
                                float* __restrict__ blk, int ld, long E) {
  long t = (long)blockIdx.x * blockDim.x + threadIdx.x;
  long e = t >> 4;
  if (e >= E) return;
  int h = (int)(t & 15) * 4;
  long r = (long)row[e], c = (long)col[e];
  float s = norm[e];
  const float4 v = *(const float4*)(hw + r * 64 + h);
  float* dst = blk + c * (long)ld + h;
  atomicAdd(dst + 0, s * v.x);
  atomicAdd(dst + 1, s * v.y);
  atomicAdd(dst + 2, s * v.z);
  atomicAdd(dst + 3, s * v.w);
}

// finalize: blk = relu(blk + dis[n]^2 * hw + bias)   (self-loop term)
__global__ void k_finalize_block(float* __restrict__ blk, int ld,
                                 const float* __restrict__ hw,
                                 const float* __restrict__ dis,
                                 const float* __restrict__ bias, long N) {
  long t = (long)blockIdx.x * blockDim.x + threadIdx.x;
  if (t >= N * 64) return;
  long n = t >> 6; int h = (int)(t & 63);
  float d = dis[n];
  float v = blk[n * (long)ld + h] + d * d * hw[n * 64 + h] + bias[h];
  blk[n * (long)ld + h] = fmaxf(v, 0.0f);
}

// ------------------------------------------------------------------
// WMMA fp32 GEMM:  Out[nrows, ncols] = A[nrows, Kd] * W[Kd, ncols]
//   - 256 threads = 8 waves; wave w owns rows [(bx*8+w)*16, +16)
//   - W staged in LDS in a pair-interleaved layout so each lane's
//     v2f B operand is ONE aligned ds_load_b64:
//       Wp[(k/2)*PRS + n*2 + (k&1)] = W[k][n]   (PRS = 160 floats;
//     160 % 64 == 32 -> lanes 0..15 hit banks 0..31, lanes 16..31 hit
//     banks 32..63: conflict-free)
//   - NT==4 path (64-col weights) stages raw W via the CDNA5 async
//     global->LDS DMA (ASYNCcnt) when available, then rearranges in LDS.
//   - V_WMMA_F32_16X16X4_F32 chained over Kd in steps of 4.
//   A layout (ISA 7.12.2, 32-bit A 16x4): lane L -> m = L%16,
//     kpair = (L/16)*2; VGPR0 = A[m][k0+kpair], VGPR1 = A[m][k0+kpair+1]
//   B layout (mirror, 4x16): lane L -> n = L%16; rows kpair, kpair+1
//   C/D layout: VGPR r -> M = r + (L/16)*8, N = L%16
// ------------------------------------------------------------------
#define PRS 160  // pair-row stride in floats

template <int NT, bool BIAS, bool RELU>
__global__ void __launch_bounds__(256)
k_wmma_gemm(const float* __restrict__ A, int lda,
            const float* __restrict__ W, int Kd,
            const float* __restrict__ bias,
            float* __restrict__ Out, int ldo,
            int ncols, long nrows) {
  extern __shared__ float lds[];
  float* Wp = lds;                         // (Kd/2)*PRS floats, swizzled
  const int tid = threadIdx.x;

  if (NT == 4) {
    // ncols == 64 here by construction: contiguous raw stage, then swizzle.
    float* raw = lds + (Kd >> 1) * PRS;    // Kd*64 floats
#if defined(__gfx1250__) && __has_builtin(__builtin_amdgcn_global_load_async_to_lds_b128)
    for (int i = tid * 4; i < Kd * 64; i += 256 * 4) {
      __builtin_amdgcn_global_load_async_to_lds_b128(
          (__attribute__((address_space(1))) v4i*)(W + i),
          (__attribute__((address_space(3))) v4i*)(raw + i),
          /*offset=*/0, /*cpol=*/0);
    }
    asm volatile("s_wait_asynccnt 0x0" ::: "memory");
#else
    for (int i = tid; i < Kd * 64; i += 256) raw[i] = W[i];
#endif
    __syncthreads();
    for (int i = tid; i < (Kd >> 1) * 128; i += 256) {
      int pr = i >> 7, rem = i & 127;
      int n = rem >> 1, sub = rem & 1;
      Wp[pr * PRS + n * 2 + sub] = raw[(pr * 2 + sub) * 64 + n];
    }
  } else {
    // classifier path: gather directly from global with zero padding
    for (int i = tid; i < (Kd >> 1) * 128; i += 256) {
      int pr = i >> 7, rem = i & 127;
      int n = rem >> 1, sub = rem & 1;
      Wp[pr * PRS + n * 2 + sub] =
          (n < ncols) ? W[(pr * 2 + sub) * ncols + n] : 0.0f;
    }
  }
  __syncthreads();

  const int wave = tid >> 5;
  const int lane = tid & 31;
  const long rowBase = ((long)blockIdx.x * 8 + wave) * 16;
  if (rowBase >= nrows) return;  // wave-uniform exit (EXEC stays all-1s)

  const int m  = lane & 15;
  const int kp = (lane >> 4) << 1;                      // 0 or 2
  const float* Arow = A + (rowBase + m) * (long)lda + kp;
  const float* Bp   = Wp + (lane >> 4) * PRS + m * 2;   // per-lane B base

  v8f acc[NT];
#pragma unroll
  for (int j = 0; j < NT; ++j) acc[j] = (v8f){};

  for (int k0 = 0; k0 < Kd; k0 += 4) {
    v2f a = *(const v2f*)(Arow + k0);                   // global_load_b64
    const float* b0 = Bp + (k0 >> 1) * PRS;
#pragma unroll
    for (int j = 0; j < NT; ++j) {
      v2f b = *(const v2f*)(b0 + j * 32);               // ds_load_b64, imm offs
      acc[j] = __builtin_amdgcn_wmma_f32_16x16x4_f32(
          /*neg_a=*/false, a, /*neg_b=*/false, b,
          /*c_mod=*/(short)0, acc[j], /*reuse_a=*/false, /*reuse_b=*/false);
    }
  }

  const int rofs = (lane >> 4) << 3;  // 0 or 8
#pragma unroll
  for (int j = 0; j < NT; ++j) {
    int cidx = j * 16 + m;
    if (cidx < ncols) {
      float bv = BIAS ? bias[cidx] : 0.0f;
#pragma unroll
      for (int r = 0; r < 8; ++r) {
        float v = acc[j][r] + bv;
        if (RELU) v = fmaxf(v, 0.0f);
        Out[(rowBase + r + rofs) * (long)ldo + cidx] = v;
      }
    }
  }
}

// ------------------------------------------------------------------
// Host launcher
// ------------------------------------------------------------------
static inline long cdivl(long a, long b) { return (a + b - 1) / b; }

extern "C" void kernel_launch(void* const* d_in, const int* in_sizes, int n_in,
                              void* d_out, int out_size, void* d_ws, size_t ws_size,
                              hipStream_t stream) {
  const float*   x      = (const float*)d_in[0];
  const int64_t* ei     = (const int64_t*)d_in[1];  // [2, E] int64 per reference
  const float*   W_ego  = (const float*)d_in[2];
  const float*   b_ego  = (const float*)d_in[3];
  const float*   W_conv = (const float*)d_in[4];    // [K, 64, 64]
  const float*   b_conv = (const float*)d_in[5];    // [K, 64]
  const float*   W_cls  = (const float*)d_in[6];    // [H*(K+1), C]
  const float*   b_cls  = (const float*)d_in[7];

  const int  D = 64, H = 64;
  const long N = (long)in_sizes[0] / D;
  const long E = (long)in_sizes[1] / 2;
  const int  K = in_sizes[5] / H;
  const int  C = in_sizes[7];
  const int  HK = H * (K + 1);  // 192

  const int64_t* rowI = ei;
  const int64_t* colI = ei + E;

  // workspace layout (floats): dis[N] | norm[E] | hw[N*H] | feat[N*HK]
  float* dis   = (float*)d_ws;
  float* normv = dis + N;
  float* hw    = normv + E;
  float* feat  = hw + N * (long)H;

  const int TB = 256;

  // ---- GCN normalization ----
  k_fill_f32<<<(unsigned)cdivl(N, TB), TB, 0, stream>>>(dis, 1.0f, N);  // self loop
  k_count_deg<<<(unsigned)cdivl(E, TB), TB, 0, stream>>>(colI, dis, E);
  k_rsqrt_inplace<<<(unsigned)cdivl(N, TB), TB, 0, stream>>>(dis, N);
  k_edge_norm<<<(unsigned)cdivl(E, TB), TB, 0, stream>>>(rowI, colI, dis, normv, E);

  const long rowTiles = N / 16;                 // N = 100000 -> 6250
  const unsigned gemmGrid = (unsigned)cdivl(rowTiles, 8);
  // LDS: swizzled pairs (+ raw staging area for the 64-col path)
  const size_t shm64  = (size_t)((D >> 1) * PRS + D * 64) * sizeof(float);   // 36 KB
  const size_t shmCls = (size_t)((HK >> 1) * PRS) * sizeof(float);           // 60 KB

  // ---- ego: feat[:,0:64] = relu(x @ W_ego + b_ego) ----
  k_wmma_gemm<4, true, true><<<gemmGrid, TB, shm64, stream>>>(
      x, D, W_ego, D, b_ego, feat, HK, H, N);

  // ---- K GCN hops ----
  for (int k = 0; k < K; ++k) {
    const float* Wk  = W_conv + (long)k * D * H;
    const float* bk  = b_conv + (long)k * H;
    const float* hIn = feat + (long)k * H;        // previous block, stride HK
    float* blkOut    = feat + (long)(k + 1) * H;  // next block, stride HK

    // hw = hIn @ Wk  (no bias / no relu yet)
    k_wmma_gemm<4, false, false><<<gemmGrid, TB, shm64, stream>>>(
        hIn, HK, Wk, D, nullptr, hw, H, H, N);

    // scatter-add over edges into blkOut
    k_zero_block<<<(unsigned)cdivl(N * 64, TB), TB, 0, stream>>>(blkOut, HK, N);
    k_scatter_edges<<<(unsigned)cdivl(E * 16, TB), TB, 0, stream>>>(
        rowI, colI, normv, hw, blkOut, HK, E);

    // + self-loop term + bias, relu
    k_finalize_block<<<(unsigned)cdivl(N * 64, TB), TB, 0, stream>>>(
        blkOut, HK, hw, dis, bk, N);
  }

  // ---- classifier: d_out = feat @ W_cls + b_cls  (C = 40 -> 3 col tiles) ----
  k_wmma_gemm<3, true, false><<<gemmGrid, TB, shmCls, stream>>>(
      feat, HK, W_cls, HK, b_cls, (float*)d_out, C, C, N);
}